// IWsoftCrossEntropy_2508260901111
// MI455X (gfx1250) — compile-verified
//
#include <hip/hip_runtime.h>
#include <cstdint>

// ---------------------------------------------------------------------------
// IW soft cross-entropy, MI455X (gfx1250).
// Pass 1 (bandwidth-bound, ~319 MB streamed once => ~13.7us floor @23.3TB/s):
//   4 pixels/thread; logits via nontemporal global_load_b128, target via
//   global_load_async_to_lds_b128 (ASYNCcnt path) consumed after exp/log
//   work. Per pixel: argmax_c, lse, contrib = lse*St - D. Block-local LDS
//   hist/acc (ds_add_f32), then 19+19 global f32 atomics per block.
// Pass 2 (1 wave): weight_table = 1/max(hist^0.2 * total^0.8, 1),
//   loss = sum_n diag(W x Acc^T)[n] via v_wmma_f32_16x16x4_f32 (5 K-chunks),
//   operands zero-padded to 16x20 in LDS so all ds loads are unconditional.
// ---------------------------------------------------------------------------

#define NCLS 19
#define NIMG 4
#define HW   (512 * 1024)
#define BLK  256
#define PPT  4                              /* pixels per thread (float4)   */
#define BLOCKS_PER_IMG (HW / (BLK * PPT))   /* 512                          */

typedef float v2f __attribute__((ext_vector_type(2)));
typedef float v4f __attribute__((ext_vector_type(4)));
typedef float v8f __attribute__((ext_vector_type(8)));

__global__ __launch_bounds__(BLK) void iw_sce_zero(float* ws) {
  int i = threadIdx.x;
  if (i < 2 * NIMG * NCLS) ws[i] = 0.0f;
}

__global__ __launch_bounds__(BLK) void iw_sce_pass1(
    const float* __restrict__ xin, const float* __restrict__ tgt,
    float* __restrict__ g_hist, float* __restrict__ g_acc) {
  __shared__ float s_hist[NCLS];
  __shared__ float s_acc[NCLS];
  __shared__ v4f   s_t[NCLS * BLK];   // [c][tid] staged target, 77824 B

  const int tid = threadIdx.x;
  const int bx  = blockIdx.x;
  const int n   = bx >> 9;                          // 512 blocks per image
  const int pix = (((bx & 511) << 8) | tid) << 2;   // first of 4 pixels
  const size_t base = (size_t)n * NCLS * HW + (size_t)pix;

  if (tid < NCLS) { s_hist[tid] = 0.0f; s_acc[tid] = 0.0f; }

  // ---- async global->LDS prefetch of 19 x float4 target values (b128).
  // GVS mode: saddr(u64 base) + vaddr(i32 byte offset); max offset ~159MB.
  {
    const uint32_t lds0 = (uint32_t)(uintptr_t)(&s_t[tid]);
    const uint64_t tb   = (uint64_t)(uintptr_t)tgt;
#pragma unroll
    for (int c = 0; c < NCLS; ++c) {
      uint32_t ldsa = lds0 + (uint32_t)(c * BLK) * 16u;
      uint32_t goff = (uint32_t)((base + (size_t)c * HW) * sizeof(float));
      asm volatile("global_load_async_to_lds_b128 %0, %1, %2"
                   :
                   : "v"(ldsa), "v"(goff), "s"(tb)
                   : "memory");
    }
  }

  // ---- logits: stream once (b128, non-temporal); per-pixel max/argmax.
  v4f   xv[NCLS];
  float vmax[PPT];
  int   amax[PPT];
#pragma unroll
  for (int p = 0; p < PPT; ++p) { vmax[p] = -3.402823466e38f; amax[p] = 0; }
#pragma unroll
  for (int c = 0; c < NCLS; ++c) {
    xv[c] = __builtin_nontemporal_load(
        (const v4f*)(xin + base + (size_t)c * HW));
#pragma unroll
    for (int p = 0; p < PPT; ++p) {
      const float x  = xv[c][p];
      const bool  gt = x > vmax[p];            // strict > == jnp first-max
      amax[p] = gt ? c : amax[p];
      vmax[p] = gt ? x : vmax[p];
    }
  }
  float lse[PPT];
#pragma unroll
  for (int p = 0; p < PPT; ++p) {
    float se = 0.0f;
#pragma unroll
    for (int c = 0; c < NCLS; ++c) se += __expf(xv[c][p] - vmax[p]);
    lse[p] = vmax[p] + __logf(se);
  }

  // ---- consume prefetched target (wave-scoped async completion).
  asm volatile("s_wait_asynccnt 0x0" ::: "memory");
  float st[PPT] = {0.0f, 0.0f, 0.0f, 0.0f};
  float dd[PPT] = {0.0f, 0.0f, 0.0f, 0.0f};
#pragma unroll
  for (int c = 0; c < NCLS; ++c) {
    const v4f tv = s_t[c * BLK + tid];         // ds_load_b128
#pragma unroll
    for (int p = 0; p < PPT; ++p) {
      const bool m = (tv[p] != -1.0f);          // IGNORE_INDEX mask
      st[p] += m ? tv[p] : 0.0f;
      dd[p] += m ? tv[p] * xv[c][p] : 0.0f;
    }
  }

  __syncthreads();   // s_hist/s_acc init visible
#pragma unroll
  for (int p = 0; p < PPT; ++p) {
    const float contrib = lse[p] * st[p] - dd[p];
    __hip_atomic_fetch_add(&s_hist[amax[p]], 1.0f, __ATOMIC_RELAXED,
                           __HIP_MEMORY_SCOPE_WORKGROUP);
    __hip_atomic_fetch_add(&s_acc[amax[p]], contrib, __ATOMIC_RELAXED,
                           __HIP_MEMORY_SCOPE_WORKGROUP);
  }
  __syncthreads();
  if (tid < NCLS) {
    __hip_atomic_fetch_add(&g_hist[n * NCLS + tid], s_hist[tid],
                           __ATOMIC_RELAXED, __HIP_MEMORY_SCOPE_AGENT);
    __hip_atomic_fetch_add(&g_acc[n * NCLS + tid], s_acc[tid],
                           __ATOMIC_RELAXED, __HIP_MEMORY_SCOPE_AGENT);
  }
}

// One wave. loss = sum_n (W x Acc^T)[n][n]. W,Acc zero-padded to 16x20 in LDS
// so every per-lane operand fetch is an unconditional, aligned ds_load_b64.
// A 16x4 f32 layout: lanes 0-15 hold K=0,1 in v0,v1; lanes 16-31 hold K=2,3.
// B 4x16 f32 layout: lanes 0-15 = N with K0/K1; lanes 16-31 = N with K2/K3.
// C/D 16x16 f32: vgpr r, lane l<16 -> (M=r, N=l).
__global__ __launch_bounds__(32) void iw_sce_pass2(
    const float* __restrict__ g_hist, const float* __restrict__ g_acc,
    float* __restrict__ out) {
  __shared__ float s_w[16 * 20];
  __shared__ float s_a[16 * 20];
  const int tid = threadIdx.x;

  for (int i = tid; i < 16 * 20; i += 32) { s_w[i] = 0.0f; s_a[i] = 0.0f; }
  __syncthreads();
  for (int i = tid; i < NIMG * NCLS; i += 32) {
    const int n = i / NCLS, c = i % NCLS;
    s_w[n * 20 + c] = g_hist[i];
    s_a[n * 20 + c] = g_acc[i];
  }
  __syncthreads();
  if (tid < NIMG) {
    float tot = 0.0f;
    for (int c = 0; c < NCLS; ++c) tot += s_w[tid * 20 + c];
    const float tp = __powf(tot, 0.8f);
    for (int c = 0; c < NCLS; ++c) {
      float d = __powf(s_w[tid * 20 + c], 0.2f) * tp;
      d = fmaxf(d, 1.0f);
      s_w[tid * 20 + c] = 1.0f / d;   // inverse-frequency weight
    }
  }
  __syncthreads();

  const int m  = tid & 15;                 // M (for A) == N (for B) per lane
  const int kb = (tid >= 16) ? 2 : 0;      // hi half-wave holds K2/K3
  v8f d8 = {};
#pragma unroll
  for (int k = 0; k < 5; ++k) {            // K = 20 in chunks of 4
    const int idx = m * 20 + 4 * k + kb;   // element index (even -> 8B align)
    const v2f av = *(const v2f*)&s_w[idx];
    const v2f bv = *(const v2f*)&s_a[idx]; // B[c][j] = acc[j][c]
    d8 = __builtin_amdgcn_wmma_f32_16x16x4_f32(false, av, false, bv,
                                               (short)0, d8, false, false);
  }
  // diagonal D[n][n]: vgpr n, lane n (n < 8 -> lanes 0-15 half)
  float loss = 0.0f;
#pragma unroll
  for (int nn = 0; nn < NIMG; ++nn) loss += __shfl(d8[nn], nn, 32);
  if (tid == 0) out[0] = loss * (1.0f / (NIMG * NCLS));
}

extern "C" void kernel_launch(void* const* d_in, const int* in_sizes, int n_in,
                              void* d_out, int out_size, void* d_ws, size_t ws_size,
                              hipStream_t stream) {
  (void)in_sizes; (void)n_in; (void)out_size; (void)ws_size;
  const float* xin = (const float*)d_in[0];
  const float* tgt = (const float*)d_in[1];
  float* out    = (float*)d_out;
  float* ws     = (float*)d_ws;
  float* g_hist = ws;                   // 76 floats
  float* g_acc  = ws + NIMG * NCLS;     // 76 floats

  iw_sce_zero<<<1, BLK, 0, stream>>>(ws);
  iw_sce_pass1<<<NIMG * BLOCKS_PER_IMG, BLK, 0, stream>>>(xin, tgt, g_hist, g_acc);
  iw_sce_pass2<<<1, 32, 0, stream>>>(g_hist, g_acc, out);
}